// TE_SFNN_3332894622254
// MI455X (gfx1250) — compile-verified
//
#include <hip/hip_runtime.h>

typedef __attribute__((ext_vector_type(2))) float v2f;
typedef __attribute__((ext_vector_type(8))) float v8f;

#define S_LEN   1000
#define B_DIM   64
#define F_DIM   512
#define H_DIM   512
#define T_DIM   1000
#define THRESH0 0.3f
#define BETA    0.02f
#define DECAY   0.5f

// -----------------------------------------------------------------------------
// Kernel 1: xh[m,h] = sum_k feat[m,k] * W[h,k] + bias[h], m = s*B + b (64000 rows)
// fp32 WMMA 16x16x4. Block = 256 threads = 8 waves. One 16-row M strip per
// block; wave w covers N columns [w*64, w*64+64) as 4 16x16 tiles.
// A frag (16x4 f32): lane l -> row m0+(l&15), K pair k + 2*(l>>4)   (b64 load)
// B frag (4x16 f32): B[k][n] = W[n][k] -> lane l reads W[n0+(l&15)][k+2*(l>>4)] (b64 load)
// C/D (16x16 f32, 8 VGPRs): vgpr r -> row m0 + r + 8*(l>>4), col n0 + (l&15)
// -----------------------------------------------------------------------------
__global__ __launch_bounds__(256) void gemm_i2h_wmma(
    const float* __restrict__ feat, const float* __restrict__ W,
    const float* __restrict__ bias, float* __restrict__ xh)
{
    const int lane  = threadIdx.x & 31;
    const int wave  = threadIdx.x >> 5;
    const int m0    = blockIdx.x << 4;     // 16-row strip
    const int lrow  = lane & 15;
    const int khalf = lane >> 4;           // 0 or 1

    const float* aptr = feat + (size_t)(m0 + lrow) * F_DIM + 2 * khalf;

    const int n0 = wave * 64;
    const float* bptr[4];
#pragma unroll
    for (int t = 0; t < 4; ++t)
        bptr[t] = W + (size_t)(n0 + t * 16 + lrow) * F_DIM + 2 * khalf;

    v8f acc[4] = {};

    for (int k = 0; k < F_DIM; k += 4) {
        v2f a = *(const v2f*)(aptr + k);
#pragma unroll
        for (int t = 0; t < 4; ++t) {
            v2f bm = *(const v2f*)(bptr[t] + k);
            acc[t] = __builtin_amdgcn_wmma_f32_16x16x4_f32(
                false, a, false, bm, (short)0, acc[t], false, false);
        }
    }

#pragma unroll
    for (int t = 0; t < 4; ++t) {
        const int n  = n0 + t * 16 + lrow;
        const float bv = bias[n];
#pragma unroll
        for (int r = 0; r < 8; ++r) {
            const int row = m0 + r + 8 * khalf;
            xh[(size_t)row * H_DIM + n] = acc[t][r] + bv;
        }
    }
}

// -----------------------------------------------------------------------------
// Kernel 2: transpose ce[H=512, T=1000] -> ce_seq[S=1000, H=512] (first S cols)
// 32x32 LDS tiles, 256 threads.
// -----------------------------------------------------------------------------
__global__ __launch_bounds__(256) void transpose_ce(
    const float* __restrict__ ce, float* __restrict__ ce_seq)
{
    __shared__ float tile[32][33];
    const int s0 = blockIdx.x * 32;
    const int h0 = blockIdx.y * 32;
    const int tx = threadIdx.x & 31;
    const int ty = threadIdx.x >> 5;   // 8 rows per pass

#pragma unroll
    for (int i = 0; i < 32; i += 8) {
        const int h = h0 + ty + i;
        const int s = s0 + tx;
        tile[ty + i][tx] = (s < S_LEN) ? ce[(size_t)h * T_DIM + s] : 0.0f;
    }
    __syncthreads();
#pragma unroll
    for (int i = 0; i < 32; i += 8) {
        const int s = s0 + ty + i;
        const int h = h0 + tx;
        if (s < S_LEN)
            ce_seq[(size_t)s * H_DIM + h] = tile[tx][ty + i];
    }
}

// -----------------------------------------------------------------------------
// Kernel 3: the scan. Elementwise recurrence per (b,h): one thread per element,
// loop over s in registers. `out` holds xh in [0, S*B*H) on entry; each thread
// reads xh[s,b,h] and overwrites it with spike[s,b,h] (same address, read
// before write, per-thread -> safe in-place). Tail: spike, mem, thresh, step.
// ce indexed via strides so either [S,H] (transposed, coalesced) or raw [H,T]
// layouts work with one kernel.
// -----------------------------------------------------------------------------
__global__ __launch_bounds__(256) void te_sfnn_scan(
    const float* __restrict__ ce, long ce_stride_s, long ce_stride_h,
    float* out)
{
    const int idx = blockIdx.x * 256 + threadIdx.x;   // 0 .. B*H-1
    const int h = idx & (H_DIM - 1);
    const size_t step = (size_t)B_DIM * H_DIM;

    float spike = 0.0f;
    float mem   = 0.0f;
    float th    = THRESH0;

    float* p = out + (size_t)idx;                     // idx = b*H + h
    const float* cep = ce + (size_t)h * ce_stride_h;

#pragma unroll 4
    for (int s = 0; s < S_LEN; ++s) {
        if (s + 8 < S_LEN)
            __builtin_prefetch(p + (size_t)(s + 8) * step, 0, 0);
        const float ce_t = cep[(size_t)s * ce_stride_s];
        const float x_t  = p[(size_t)s * step];
        th   = th + mem * ce_t - (th - THRESH0) * BETA;
        mem  = mem * DECAY * (1.0f - spike) + x_t;
        spike = (mem > th) ? 1.0f : 0.0f;
        p[(size_t)s * step] = spike;
    }

    float* tail = out + (size_t)S_LEN * step;
    tail[idx]            = spike;
    tail[step + idx]     = mem;
    tail[2 * step + idx] = th;
    if (idx == 0)
        tail[3 * step] = (float)S_LEN;   // step_num
}

extern "C" void kernel_launch(void* const* d_in, const int* in_sizes, int n_in,
                              void* d_out, int out_size, void* d_ws, size_t ws_size,
                              hipStream_t stream) {
    (void)in_sizes; (void)n_in; (void)out_size;
    const float* feat = (const float*)d_in[0];   // [S,B,F]
    const float* W    = (const float*)d_in[1];   // [H,F]
    const float* bias = (const float*)d_in[2];   // [H]
    const float* ce   = (const float*)d_in[3];   // [H,T]
    float* out = (float*)d_out;

    // Phase 1: GEMM -> xh written into the out_seq region of d_out (in-place reuse)
    const int m_tiles = (S_LEN * B_DIM) / 16;    // 4000
    gemm_i2h_wmma<<<m_tiles, 256, 0, stream>>>(feat, W, bias, out);

    // Phase 2+3: ce transpose (if workspace fits) then the elementwise scan
    const size_t ce_seq_bytes = (size_t)S_LEN * H_DIM * sizeof(float);
    const int scan_blocks = (B_DIM * H_DIM) / 256;   // 128
    if (ws_size >= ce_seq_bytes) {
        float* ce_seq = (float*)d_ws;
        dim3 tgrid((S_LEN + 31) / 32, H_DIM / 32);
        transpose_ce<<<tgrid, 256, 0, stream>>>(ce, ce_seq);
        te_sfnn_scan<<<scan_blocks, 256, 0, stream>>>(ce_seq, (long)H_DIM, 1L, out);
    } else {
        te_sfnn_scan<<<scan_blocks, 256, 0, stream>>>(ce, 1L, (long)T_DIM, out);
    }
}